// STGNN_29927332119080
// MI455X (gfx1250) — compile-verified
//
#include <hip/hip_runtime.h>
#include <hip/hip_bf16.h>
#include <math.h>

// ---------------------------------------------------------------------------
// STGNN forward for MI455X (gfx1250, wave32).
// All GEMMs use V_WMMA_F32_16X16X4_F32 (fp32 matrix cores, RNE) since the
// reference is fp32 and the 65-GFLOP workload is L2-resident / latency bound.
// ---------------------------------------------------------------------------

#define B_ 8
#define T_ 12
#define N_ 1024
#define F_ 64
#define H_ 4
#define D_ 16
#define BTN_ (B_ * T_ * N_)   // 98304 rows of F_ features

typedef float v2f __attribute__((ext_vector_type(2)));
typedef float v8f __attribute__((ext_vector_type(8)));

__device__ __forceinline__ v8f wmma_f32(v2f a, v2f b, v8f c) {
  // 8 args: (neg_a, A, neg_b, B, c_mod, C, reuse_a, reuse_b)
  return __builtin_amdgcn_wmma_f32_16x16x4_f32(false, a, false, b, (short)0, c,
                                               false, false);
}

// ---------------------------------------------------------------------------
// Generic batched GEMM:  Y[m,n] = act( rowscale[m] * (X @ W)[m,n] + bias[n] )
//   X: M x K row-major (ldx), per-batch stride sxb
//   W: K x Nout (ldw) if WTRANS==0, else W[n,k] access (gram: B = Xt^T)
//   block = 256 threads = 8 waves; block tile = 128 rows x 64 cols
//   wave tile = 16 rows x 64 cols = 4 WMMA C tiles; K stepped by 8
//   B chunk (8 x 64) staged in LDS, shared by all 8 waves -> 8 WMMA per
//   barrier pair.  WTRANS/ACT are compile-time so the K loop is branch-free.
// M % 128 == 0, Nout % 64 == 0, K % 8 == 0 for all call sites.
// ACT: 0=none 1=relu 2=sigmoid 3=tanh
// ---------------------------------------------------------------------------
template <int WTRANS, int ACT>
__global__ __launch_bounds__(256) void gemm_wmma_kernel(
    const float* __restrict__ X, const float* __restrict__ W,
    const float* __restrict__ bias, const float* __restrict__ rowscale,
    float* __restrict__ Y, int M, int K, int Nout, int ldx, int ldw, int ldy,
    long sxb, long swb, long syb, long srb) {
  __shared__ float Bs[8][64];

  const int lane = threadIdx.x & 31;
  const int wv   = threadIdx.x >> 5;
  const int hi   = lane >> 4;   // 0 | 1 (half-wave)
  const int lo   = lane & 15;
  const int m0   = blockIdx.x * 128 + wv * 16;
  const int n0   = blockIdx.y * 64;
  const int bat  = blockIdx.z;

  const float* Xb = X + (long)bat * sxb;
  const float* Wb = W + (long)bat * swb;
  float*       Yb = Y + (long)bat * syb;

  v8f acc0 = {0,0,0,0,0,0,0,0};
  v8f acc1 = acc0, acc2 = acc0, acc3 = acc0;

  const int kk = threadIdx.x >> 6;   // 0..3 (staging role)
  const int nn = threadIdx.x & 63;

  for (int k0 = 0; k0 < K; k0 += 8) {
    // stage B chunk (8 x 64) into LDS
    if (WTRANS) {
      const float* wr = Wb + (long)(n0 + nn) * ldw + k0 + kk;
      Bs[kk][nn]     = wr[0];
      Bs[kk + 4][nn] = wr[4];
    } else {
      const float* wr = Wb + (long)(k0 + kk) * ldw + (n0 + nn);
      Bs[kk][nn]     = wr[0];
      Bs[kk + 4][nn] = wr[(long)4 * ldw];
    }
    __syncthreads();

    // A fragments: 16x4, M = lo, K = k0 + {0,4} + 2*hi + {0,1}
    const float* xr = Xb + (long)(m0 + lo) * ldx + k0 + 2 * hi;
    v2f a0, a1;
    a0.x = xr[0];
    a0.y = xr[1];
    a1.x = xr[4];
    a1.y = xr[5];
    __builtin_prefetch(xr + 8, 0, 1);  // speculative; silently dropped OOB

#pragma unroll
    for (int half = 0; half < 2; ++half) {
      const v2f a = half ? a1 : a0;
      const int kb = half * 4 + 2 * hi;
      {
        v2f b = {Bs[kb][lo], Bs[kb + 1][lo]};
        acc0 = wmma_f32(a, b, acc0);
      }
      {
        v2f b = {Bs[kb][16 + lo], Bs[kb + 1][16 + lo]};
        acc1 = wmma_f32(a, b, acc1);
      }
      {
        v2f b = {Bs[kb][32 + lo], Bs[kb + 1][32 + lo]};
        acc2 = wmma_f32(a, b, acc2);
      }
      {
        v2f b = {Bs[kb][48 + lo], Bs[kb + 1][48 + lo]};
        acc3 = wmma_f32(a, b, acc3);
      }
    }
    __syncthreads();
  }

  // Epilogue.  C layout: element r -> row = r + 8*hi, col = lane&15
#pragma unroll
  for (int nt = 0; nt < 4; ++nt) {
    v8f acc = (nt == 0) ? acc0 : (nt == 1) ? acc1 : (nt == 2) ? acc2 : acc3;
#pragma unroll
    for (int r = 0; r < 8; ++r) {
      const int rr = m0 + r + 8 * hi;
      const int cc = n0 + nt * 16 + lo;
      float vv = acc[r];
      if (rowscale) vv *= rowscale[(long)bat * srb + rr];
      if (bias) vv += bias[cc];
      if (ACT == 1) vv = fmaxf(vv, 0.f);
      else if (ACT == 2) vv = 1.f / (1.f + __expf(-vv));
      else if (ACT == 3) vv = tanhf(vv);
      Yb[(long)rr * ldy + cc] = vv;
    }
  }
}

// ---------------------------------------------------------------------------
// sgnn graph row pass: in-place row softmax of R (relu(softmax)=softmax since
// softmax > 0), add identity, compute d = rowsum^-0.5 (inf -> 0).
// One wave per row of length N_=1024.
// ---------------------------------------------------------------------------
__global__ __launch_bounds__(256) void graph_row_kernel(float* __restrict__ R,
                                                        float* __restrict__ dvec) {
  const long row = (long)blockIdx.x * 8 + (threadIdx.x >> 5);  // 0 .. B*N-1
  const int lane = threadIdx.x & 31;
  const int i    = (int)(row & (N_ - 1));
  float* Rr = R + row * N_;

  float mx = -1e30f;
  for (int j = lane; j < N_; j += 32) mx = fmaxf(mx, Rr[j]);
#pragma unroll
  for (int m = 1; m < 32; m <<= 1) mx = fmaxf(mx, __shfl_xor(mx, m, 32));

  float s = 0.f;
  for (int j = lane; j < N_; j += 32) s += __expf(Rr[j] - mx);
#pragma unroll
  for (int m = 1; m < 32; m <<= 1) s += __shfl_xor(s, m, 32);
  const float inv = 1.f / s;

  float rs = 0.f;
  for (int j = lane; j < N_; j += 32) {
    float val = __expf(Rr[j] - mx) * inv;  // == relu(softmax)
    if (j == i) val += 1.f;                // + eye
    Rr[j] = val;
    rs += val;
  }
#pragma unroll
  for (int m = 1; m < 32; m <<= 1) rs += __shfl_xor(rs, m, 32);

  if (lane == 0) {
    float d = (rs > 0.f) ? rsqrtf(rs) : 0.f;  // where(isinf(d), 0, d)
    dvec[row] = d;
  }
}

// xs[b,n,f] = dvec[b*N+n] * src[b*sxb + n*F + f]
__global__ void scale_rows_kernel(const float* __restrict__ src, long sxb,
                                  const float* __restrict__ dvec,
                                  float* __restrict__ dst) {
  const long i = (long)blockIdx.x * blockDim.x + threadIdx.x;  // B*N*F
  const long b = i / ((long)N_ * F_);
  const long rem = i - b * (long)N_ * F_;
  dst[i] = dvec[i >> 6] * src[b * sxb + rem];
}

// embed: xe[b,t,n,f] = x[b,t,n] * start_w[f] + start_b[f]
__global__ void embed_kernel(const float* __restrict__ x,
                             const float* __restrict__ sw,
                             const float* __restrict__ sb,
                             float* __restrict__ xe) {
  const long i = (long)blockIdx.x * blockDim.x + threadIdx.x;  // B*T*N*F
  const int f = (int)(i & (F_ - 1));
  xe[i] = x[i >> 6] * sw[f] + sb[f];
}

__global__ void zero_kernel(float* __restrict__ p) {
  const long i = (long)blockIdx.x * blockDim.x + threadIdx.x;
  p[i] = 0.f;
}

// cat = [a | b]  (rows of 2F)
__global__ void concat_kernel(const float* __restrict__ a,
                              const float* __restrict__ b,
                              float* __restrict__ dst) {
  const long i = (long)blockIdx.x * blockDim.x + threadIdx.x;  // B*N*2F
  const long row = i >> 7;
  const int c = (int)(i & 127);
  dst[i] = (c < F_) ? a[row * F_ + c] : b[row * F_ + (c - F_)];
}

// cat2 = [gx | r * gh],  r = ru[:, :F]
__global__ void cat2_kernel(const float* __restrict__ gx,
                            const float* __restrict__ ru,
                            const float* __restrict__ gh,
                            float* __restrict__ dst) {
  const long i = (long)blockIdx.x * blockDim.x + threadIdx.x;  // B*N*2F
  const long row = i >> 7;
  const int c = (int)(i & 127);
  if (c < F_) dst[i] = gx[row * F_ + c];
  else {
    const int f = c - F_;
    dst[i] = ru[row * 2 * F_ + f] * gh[row * F_ + f];
  }
}

// hs = u*z + (1-u)*gh ;  also write into xe[:, t]   (u = ru[:, F:])
__global__ void blend_kernel(const float* __restrict__ ru,
                             const float* __restrict__ z,
                             const float* __restrict__ gh,
                             float* __restrict__ hs, float* __restrict__ xe_t,
                             long sxb) {
  const long i = (long)blockIdx.x * blockDim.x + threadIdx.x;  // B*N*F
  const long row = i >> 6;
  const int f = (int)(i & (F_ - 1));
  const float u = ru[row * 2 * F_ + F_ + f];
  const float h = u * z[i] + (1.f - u) * gh[i];
  hs[i] = h;
  const long b = i / ((long)N_ * F_);
  const long rem = i - b * (long)N_ * F_;
  xe_t[b * sxb + rem] = h;
}

// xe += positional encoding
__global__ void peadd_kernel(float* __restrict__ xe) {
  const long i = (long)blockIdx.x * blockDim.x + threadIdx.x;  // B*T*N*F
  const int f = (int)(i & (F_ - 1));
  const int t = (int)((i / ((long)N_ * F_)) % T_);
  const int j = f >> 1;
  const float div = __expf((float)(2 * j) * (-logf(10000.f) / (float)F_));
  const float arg = (float)t * div;
  xe[i] += (f & 1) ? __cosf(arg) : __sinf(arg);
}

// ---------------------------------------------------------------------------
// Attention over T per (b, n, h): one wave per head-node, T=12 padded to 16.
// Two WMMA passes (q k^T, then att v); scores bounced through LDS to
// re-fragment C-layout -> A-layout.
// Masked fragment loads use CLAMPED addresses + value selects (v_cndmask)
// instead of predicated loads, so the loop stays free of EXEC branching.
// ---------------------------------------------------------------------------
__global__ __launch_bounds__(256) void attn_kernel(const float* __restrict__ q,
                                                   const float* __restrict__ k,
                                                   const float* __restrict__ v,
                                                   float* __restrict__ o) {
  __shared__ float att_s[8][16][17];
  const int wv = threadIdx.x >> 5;
  const int lane = threadIdx.x & 31;
  const int gw = blockIdx.x * 8 + wv;  // 0 .. B*N*H-1
  const int h = gw & (H_ - 1);
  const int n = (gw >> 2) & (N_ - 1);
  const int b = gw >> 12;
  const int hi = lane >> 4, lo = lane & 15;

  const long rstr = (long)N_ * F_;  // stride between consecutive t
  const long base = ((long)(b * T_) * N_ + n) * F_ + h * D_;  // (t=0, d=0)

  // ---- att = (q k^T) / sqrt(D), rows t = lo (clamped), K = d
  const bool rvalid = (lo < T_);
  const int tcl = rvalid ? lo : 0;  // clamped row -> loads always in-bounds
  const float* qrow = q + base + tcl * rstr + 2 * hi;
  const float* krow = k + base + tcl * rstr + 2 * hi;
  const v2f zz = {0.f, 0.f};

  v8f att = {0,0,0,0,0,0,0,0};
#pragma unroll
  for (int kc = 0; kc < 4; ++kc) {
    v2f a  = *(const v2f*)(qrow + 4 * kc);   // aligned float2 (d pairs)
    v2f bb = *(const v2f*)(krow + 4 * kc);
    a = rvalid ? a : zz;                      // v_cndmask, no EXEC branch
    bb = rvalid ? bb : zz;
    att = wmma_f32(a, bb, att);
  }

  // masked row softmax over s (16-lane group reductions)
  const float scale = 0.25f;  // 1/sqrt(D)
  float vals[8];
#pragma unroll
  for (int r = 0; r < 8; ++r) {
    float x = (lo < T_) ? att[r] * scale : -1e30f;
    float mx = x;
#pragma unroll
    for (int m = 1; m < 16; m <<= 1) mx = fmaxf(mx, __shfl_xor(mx, m, 32));
    float e = (lo < T_) ? __expf(x - mx) : 0.f;
    float s = e;
#pragma unroll
    for (int m = 1; m < 16; m <<= 1) s += __shfl_xor(s, m, 32);
    vals[r] = e / s;
  }

#pragma unroll
  for (int r = 0; r < 8; ++r) att_s[wv][r + 8 * hi][lo] = vals[r];
  __syncthreads();

  // ---- o = att @ v  (K = s; att cols >= T are exactly 0 after softmax)
  v8f oacc = {0,0,0,0,0,0,0,0};
#pragma unroll
  for (int kc = 0; kc < 4; ++kc) {
    const int s0 = kc * 4 + 2 * hi;
    const int s0c = (s0 < T_) ? s0 : 0;       // clamped row indices
    const int s1c = (s0 + 1 < T_) ? s0 + 1 : 0;
    float b0 = v[base + (long)s0c * rstr + lo];
    float b1 = v[base + (long)s1c * rstr + lo];
    v2f a, bb;
    a.x = att_s[wv][lo][s0];
    a.y = att_s[wv][lo][s0 + 1];
    bb.x = (s0 < T_) ? b0 : 0.f;              // value select only
    bb.y = (s0 + 1 < T_) ? b1 : 0.f;
    oacc = wmma_f32(a, bb, oacc);
  }

#pragma unroll
  for (int r = 0; r < 8; ++r) {
    const int t = r + 8 * hi;
    if (t < T_) o[base + (long)t * rstr + lo] = oacc[r];  // EXEC-masked store
  }
}

// LayerNorm over F=64 per row, one wave per row (2 elems/lane), optional
// residual input.
__global__ __launch_bounds__(256) void ln_kernel(const float* __restrict__ x,
                                                 const float* __restrict__ res,
                                                 const float* __restrict__ g,
                                                 const float* __restrict__ be,
                                                 float* __restrict__ y) {
  const long row = (long)blockIdx.x * 8 + (threadIdx.x >> 5);
  const int lane = threadIdx.x & 31;
  float a = x[row * F_ + lane];
  float b = x[row * F_ + lane + 32];
  if (res) {
    a += res[row * F_ + lane];
    b += res[row * F_ + lane + 32];
  }
  float s = a + b;
#pragma unroll
  for (int m = 1; m < 32; m <<= 1) s += __shfl_xor(s, m, 32);
  const float mu = s * (1.f / 64.f);
  const float da = a - mu, db = b - mu;
  float vs = da * da + db * db;
#pragma unroll
  for (int m = 1; m < 32; m <<= 1) vs += __shfl_xor(vs, m, 32);
  const float inv = rsqrtf(vs * (1.f / 64.f) + 1e-5f);
  y[row * F_ + lane]      = da * inv * g[lane] + be[lane];
  y[row * F_ + lane + 32] = db * inv * g[lane + 32] + be[lane + 32];
}

// out[row] = dot(xe[row,:], end_w[:,0]) + end_b[0]; one wave per row
__global__ __launch_bounds__(256) void end_kernel(const float* __restrict__ xe,
                                                  const float* __restrict__ ew,
                                                  const float* __restrict__ eb,
                                                  float* __restrict__ out) {
  const long row = (long)blockIdx.x * 8 + (threadIdx.x >> 5);
  const int lane = threadIdx.x & 31;
  float s = xe[row * F_ + lane] * ew[lane] +
            xe[row * F_ + lane + 32] * ew[lane + 32];
#pragma unroll
  for (int m = 1; m < 32; m <<= 1) s += __shfl_xor(s, m, 32);
  if (lane == 0) out[row] = s + eb[0];
}

// ---------------------------------------------------------------------------
extern "C" void kernel_launch(void* const* d_in, const int* in_sizes, int n_in,
                              void* d_out, int out_size, void* d_ws,
                              size_t ws_size, hipStream_t stream) {
  (void)in_sizes; (void)n_in; (void)out_size; (void)ws_size;

  const float* x       = (const float*)d_in[0];
  const float* start_w = (const float*)d_in[1];
  const float* start_b = (const float*)d_in[2];
  const float* sg_w1   = (const float*)d_in[3];
  const float* sg_b1   = (const float*)d_in[4];
  const float* sg_w2   = (const float*)d_in[5];
  const float* sg_b2   = (const float*)d_in[6];
  const float* sg_w3   = (const float*)d_in[7];
  const float* sg_b3   = (const float*)d_in[8];
  const float* gru_w   = (const float*)d_in[9];
  const float* gru_b   = (const float*)d_in[10];
  const float* gruz_w  = (const float*)d_in[11];
  const float* gruz_b  = (const float*)d_in[12];
  const float* q_w     = (const float*)d_in[13];
  const float* q_b     = (const float*)d_in[14];
  const float* k_w     = (const float*)d_in[15];
  const float* k_b     = (const float*)d_in[16];
  const float* v_w     = (const float*)d_in[17];
  const float* v_b     = (const float*)d_in[18];
  const float* ln1_g   = (const float*)d_in[19];
  const float* ln1_b   = (const float*)d_in[20];
  const float* ff1_w   = (const float*)d_in[21];
  const float* ff1_b   = (const float*)d_in[22];
  const float* ff2_w   = (const float*)d_in[23];
  const float* ff2_b   = (const float*)d_in[24];
  const float* ln2_g   = (const float*)d_in[25];
  const float* ln2_b   = (const float*)d_in[26];
  const float* end_w   = (const float*)d_in[27];
  const float* end_b   = (const float*)d_in[28];

  // ---- workspace layout (floats). Total = 6.29M + 4*6.29M = 31.46M floats.
  const size_t XE = (size_t)B_ * T_ * N_ * F_;  // 6291456
  float* W   = (float*)d_ws;
  float* xe  = W;
  float* S   = W + XE;          // scratch region (25.2M floats)
  // transformer phase (after t-loop; reuses scratch region)
  float* qb_ = S;
  float* kb_ = S + XE;
  float* vb_ = S + 2 * XE;
  float* ob_ = S + 3 * XE;
  // sgnn/gru phase (overlaps transformer buffers, disjoint in time)
  const size_t BNF = (size_t)B_ * N_ * F_;      // 524288
  float* bufA = S;                              // B*N*N = 8388608
  float* p1   = bufA + (size_t)B_ * N_ * N_;
  float* p2   = p1 + BNF;
  float* gxb  = p2 + BNF;
  float* ghb  = gxb + BNF;
  float* hsb  = ghb + BNF;
  float* xsb  = hsb + BNF;
  float* axb  = xsb + BNF;
  float* dvec = axb + BNF;                      // B*N = 8192
  float* catb = dvec + (size_t)B_ * N_;         // B*N*2F
  float* rub  = catb + 2 * BNF;
  float* cat2 = rub + 2 * BNF;
  float* zb   = cat2 + 2 * BNF;

  auto gemm = [&](const float* X, long sxb, int ldx, const float* Wm, long swb,
                  int ldw, int wtrans, const float* bias, const float* rsc,
                  long srb, float* Y, long syb, int ldy, int M, int K, int Nout,
                  int act, int batch) {
    dim3 g(M / 128, Nout / 64, batch);
    dim3 blk(256);
#define GEMM_ARGS X, Wm, bias, rsc, Y, M, K, Nout, ldx, ldw, ldy, sxb, swb, syb, srb
    if (wtrans)
      hipLaunchKernelGGL((gemm_wmma_kernel<1, 0>), g, blk, 0, stream, GEMM_ARGS);
    else if (act == 0)
      hipLaunchKernelGGL((gemm_wmma_kernel<0, 0>), g, blk, 0, stream, GEMM_ARGS);
    else if (act == 1)
      hipLaunchKernelGGL((gemm_wmma_kernel<0, 1>), g, blk, 0, stream, GEMM_ARGS);
    else if (act == 2)
      hipLaunchKernelGGL((gemm_wmma_kernel<0, 2>), g, blk, 0, stream, GEMM_ARGS);
    else
      hipLaunchKernelGGL((gemm_wmma_kernel<0, 3>), g, blk, 0, stream, GEMM_ARGS);
#undef GEMM_ARGS
  };

  const long NF = (long)N_ * F_;
  const long NN = (long)N_ * N_;

  // sgnn(xin) -> dst ; tw = which*T + t selects the weight slice (l=0, L=1)
  auto sgnn = [&](const float* xin, long sxb, int tw, float* dst) {
    const float* w1 = sg_w1 + (size_t)tw * F_ * F_;
    const float* b1 = sg_b1 + (size_t)tw * F_;
    const float* w2 = sg_w2 + (size_t)tw * F_ * F_;
    const float* b2 = sg_b2 + (size_t)tw * F_;
    const float* w3 = sg_w3 + (size_t)tw * F_ * F_;
    const float* b3 = sg_b3 + (size_t)tw * F_;
    // p = (x @ w1 + b1) @ w2 + b2
    gemm(xin, sxb, F_, w1, 0, F_, 0, b1, nullptr, 0, p1, NF, F_, N_, F_, F_, 0, B_);
    gemm(p1, NF, F_, w2, 0, F_, 0, b2, nullptr, 0, p2, NF, F_, N_, F_, F_, 0, B_);
    // a = p p^T  (B-transposed WMMA GEMM)
    gemm(p2, NF, F_, p2, NF, F_, 1, nullptr, nullptr, 0, bufA, NN, N_, N_, F_, N_, 0, B_);
    // R = softmax + I ; d = rowsum^-0.5
    hipLaunchKernelGGL(graph_row_kernel, dim3((B_ * N_) / 8), dim3(256), 0,
                       stream, bufA, dvec);
    // xs = d_j * x
    hipLaunchKernelGGL(scale_rows_kernel, dim3(BNF / 256), dim3(256), 0, stream,
                       xin, sxb, dvec, xsb);
    // ax = d_i * (R @ xs)
    gemm(bufA, NN, N_, xsb, NF, F_, 0, nullptr, dvec, N_, axb, NF, F_, N_, N_,
         F_, 0, B_);
    // dst = relu(ax @ w3 + b3)
    gemm(axb, NF, F_, w3, 0, F_, 0, b3, nullptr, 0, dst, NF, F_, N_, F_, F_, 1, B_);
  };

  // ---- embed
  hipLaunchKernelGGL(embed_kernel, dim3(XE / 256), dim3(256), 0, stream, x,
                     start_w, start_b, xe);

  // ---- L=1 recurrent layer over t
  for (int t = 0; t < T_; ++t) {
    sgnn(xe + (size_t)t * NF, (long)T_ * NF, 0 * T_ + t, gxb);
    if (t == 0)
      hipLaunchKernelGGL(zero_kernel, dim3(BNF / 256), dim3(256), 0, stream, ghb);
    else
      sgnn(hsb, NF, 1 * T_ + t, ghb);
    // GRU
    hipLaunchKernelGGL(concat_kernel, dim3(2 * BNF / 256), dim3(256), 0, stream,
                       gxb, ghb, catb);
    gemm(catb, 2 * NF, 2 * F_, gru_w + (size_t)t * 4 * F_ * F_, 0, 2 * F_, 0,
         gru_b + (size_t)t * 2 * F_, nullptr, 0, rub, 2 * NF, 2 * F_, N_,
         2 * F_, 2 * F_, 2 /*sigmoid*/, B_);
    hipLaunchKernelGGL(cat2_kernel, dim3(2 * BNF / 256), dim3(256), 0, stream,
                       gxb, rub, ghb, cat2);
    gemm(cat2, 2 * NF, 2 * F_, gruz_w + (size_t)t * 2 * F_ * F_, 0, F_, 0,
         gruz_b + (size_t)t * F_, nullptr, 0, zb, NF, F_, N_, 2 * F_, F_,
         3 /*tanh*/, B_);
    hipLaunchKernelGGL(blend_kernel, dim3(BNF / 256), dim3(256), 0, stream, rub,
                       zb, ghb, hsb, xe + (size_t)t * NF, (long)T_ * NF);
  }

  // ---- positional encoding
  hipLaunchKernelGGL(peadd_kernel, dim3(XE / 256), dim3(256), 0, stream, xe);

  // ---- transformer layer (L=1)
  gemm(xe, 0, F_, q_w, 0, F_, 0, q_b, nullptr, 0, qb_, 0, F_, BTN_, F_, F_, 0, 1);
  gemm(xe, 0, F_, k_w, 0, F_, 0, k_b, nullptr, 0, kb_, 0, F_, BTN_, F_, F_, 0, 1);
  gemm(xe, 0, F_, v_w, 0, F_, 0, v_b, nullptr, 0, vb_, 0, F_, BTN_, F_, F_, 0, 1);
  hipLaunchKernelGGL(attn_kernel, dim3((B_ * N_ * H_) / 8), dim3(256), 0,
                     stream, qb_, kb_, vb_, ob_);
  // o = ln(o + xe)
  hipLaunchKernelGGL(ln_kernel, dim3(BTN_ / 8), dim3(256), 0, stream, ob_, xe,
                     ln1_g, ln1_b, ob_);
  // f = relu(o @ fw1 + fb1) @ fw2 + fb2   (q/k buffers are free now)
  gemm(ob_, 0, F_, ff1_w, 0, F_, 0, ff1_b, nullptr, 0, qb_, 0, F_, BTN_, F_, F_, 1, 1);
  gemm(qb_, 0, F_, ff2_w, 0, F_, 0, ff2_b, nullptr, 0, kb_, 0, F_, BTN_, F_, F_, 0, 1);
  // xe = ln(f + o)
  hipLaunchKernelGGL(ln_kernel, dim3(BTN_ / 8), dim3(256), 0, stream, kb_, ob_,
                     ln2_g, ln2_b, xe);

  // ---- output projection
  hipLaunchKernelGGL(end_kernel, dim3(BTN_ / 8), dim3(256), 0, stream, xe,
                     end_w, end_b, (float*)d_out);
}